// GaussianNoiseLayer_3985729650989
// MI455X (gfx1250) — compile-verified
//
#include <hip/hip_runtime.h>
#include <cstdint>

// Per-row global-norm clip + Gaussian noise add.
//   x, noise : [B, F] fp32, F = 4096, B = 16384
//   out = x * (clip / max(||x_row||, clip)) + noise
//
// One block per row. 256 threads * 16 floats = 4096 cols.
// x is held in VGPRs across the reduction (read once).
// noise is staged LDS via CDNA5 async global->LDS b128 loads (ASYNCcnt),
// overlapped with the wave32 shuffle reduction + block barrier.

#define F_COLS 4096
#define TPB    256
#define CHUNKS 4              // 4 x float4 per thread = 16 floats
#define CLIP_F 3.0f

__global__ __launch_bounds__(TPB) void dp_clip_noise_kernel(
    const float* __restrict__ x,
    const float* __restrict__ noise,
    float* __restrict__ out)
{
    __shared__ float s_noise[F_COLS];          // 16 KB staged noise row
    __shared__ float s_partial[TPB / 32];      // per-wave partial sums

    const int   t      = threadIdx.x;
    const size_t rowOff = (size_t)blockIdx.x * F_COLS;
    const float* xr = x     + rowOff;
    const float* nr = noise + rowOff;
    float*       orw = out  + rowOff;

    // ---- 1) Kick off async noise -> LDS (4 x b128 per thread) -------------
    // GVS addressing: mem = SADDR(64b sgpr) + VADDR(32b byte offset).
    // Each thread later reads back only the LDS bytes it requested, so a
    // per-wave s_wait_asynccnt 0 is the only ordering needed for this data.
    {
        const unsigned long long nbase = (unsigned long long)nr;
        const unsigned lds_base = (unsigned)(uintptr_t)(&s_noise[0]); // low 32b of flat addr == LDS byte offset
        #pragma unroll
        for (int c = 0; c < CHUNKS; ++c) {
            const unsigned byteoff = (unsigned)((c * (TPB * 4) + t * 4) * sizeof(float));
            const unsigned ldsoff  = lds_base + byteoff;
            asm volatile("global_load_async_to_lds_b128 %0, %1, %2"
                         :: "v"(ldsoff), "v"(byteoff), "s"(nbase)
                         : "memory");
        }
    }

    // ---- 2) Load x row into VGPRs (b128, coalesced), sum of squares -------
    float4 vx[CHUNKS];
    float  ss = 0.0f;
    #pragma unroll
    for (int c = 0; c < CHUNKS; ++c) {
        vx[c] = reinterpret_cast<const float4*>(xr)[c * TPB + t];
        ss = fmaf(vx[c].x, vx[c].x, ss);
        ss = fmaf(vx[c].y, vx[c].y, ss);
        ss = fmaf(vx[c].z, vx[c].z, ss);
        ss = fmaf(vx[c].w, vx[c].w, ss);
    }

    // ---- 3) wave32 butterfly reduction (gfx1250 is wave32-only) -----------
    #pragma unroll
    for (int m = 16; m >= 1; m >>= 1)
        ss += __shfl_xor(ss, m, 32);

    if ((t & 31) == 0) s_partial[t >> 5] = ss;
    __syncthreads();

    float total = 0.0f;
    #pragma unroll
    for (int w = 0; w < TPB / 32; ++w)
        total += s_partial[w];                 // broadcast LDS reads, uniform

    const float norm  = sqrtf(total);
    const float scale = CLIP_F / fmaxf(norm, CLIP_F);

    // ---- 4) Wait for our async noise tiles, then fused scale+add ----------
    asm volatile("s_wait_asynccnt 0" ::: "memory");

    #pragma unroll
    for (int c = 0; c < CHUNKS; ++c) {
        const float4 nz = reinterpret_cast<const float4*>(s_noise)[c * TPB + t];
        float4 o;
        o.x = fmaf(vx[c].x, scale, nz.x);
        o.y = fmaf(vx[c].y, scale, nz.y);
        o.z = fmaf(vx[c].z, scale, nz.z);
        o.w = fmaf(vx[c].w, scale, nz.w);
        reinterpret_cast<float4*>(orw)[c * TPB + t] = o;   // b128 store
    }
}

extern "C" void kernel_launch(void* const* d_in, const int* in_sizes, int n_in,
                              void* d_out, int out_size, void* d_ws, size_t ws_size,
                              hipStream_t stream) {
    const float* x     = (const float*)d_in[0];
    const float* noise = (const float*)d_in[1];
    float*       out   = (float*)d_out;

    const int rows = in_sizes[0] / F_COLS;     // 16384 for the reference shapes

    dp_clip_noise_kernel<<<dim3(rows), dim3(TPB), 0, stream>>>(x, noise, out);
}